// SCQ_layer_74079595922107
// MI455X (gfx1250) — compile-verified
//
#include <hip/hip_runtime.h>
#include <math.h>

// ---------------- problem constants ----------------
#define M_TOK 32768   // B*N
#define D_EMB 256
#define K_EMB 512
#define NS_ITERS 16
#define EPSV 1e-5f
#define LAMBDA_REG 1.0f
#define A_ENT 0.5f
#define A_DIV 0.5f

typedef float v2f __attribute__((ext_vector_type(2)));
typedef float v8f __attribute__((ext_vector_type(8)));

// ---------------- WMMA f32 16x16x4 core ----------------
__device__ inline v8f wmma4(v2f a, v2f b, v8f c) {
  return __builtin_amdgcn_wmma_f32_16x16x4_f32(
      /*neg_a=*/false, a, /*neg_b=*/false, b,
      /*c_mod=*/(short)0, c, /*reuse_a=*/false, /*reuse_b=*/false);
}

// AMODE 0: A[m*lda+k] row-major.   AMODE 1: A[k*lda+m] (transposed storage).
// BMODE 0: B[k*ldb+n] row-major.   BMODE 1: B[n*ldb+k] (transposed storage).
// Each wave computes a 16x64 output strip (4 accumulators); A frag reused 4x.
template <int AMODE, int BMODE>
__device__ inline void gemm_core(const float* __restrict__ A, int lda,
                                 const float* __restrict__ B, int ldb,
                                 int Kd, int m0, int n0, v8f acc[4]) {
  const int lane = threadIdx.x & 31;
  const int r = lane & 15;
  const int half = lane >> 4;
  for (int kk = 0; kk < Kd; kk += 4) {
    const int ka = kk + 2 * half;
    v2f a;
    if (AMODE == 0) {
      const float* ap = A + (size_t)(m0 + r) * lda + ka;
      a.x = ap[0]; a.y = ap[1];
    } else {
      a.x = A[(size_t)(ka + 0) * lda + (m0 + r)];
      a.y = A[(size_t)(ka + 1) * lda + (m0 + r)];
    }
#pragma unroll
    for (int t = 0; t < 4; ++t) {
      const int n = n0 + 16 * t + r;
      v2f b;
      if (BMODE == 0) {
        b.x = B[(size_t)(ka + 0) * ldb + n];
        b.y = B[(size_t)(ka + 1) * ldb + n];
      } else {
        const float* bp = B + (size_t)n * ldb + ka;
        b.x = bp[0]; b.y = bp[1];
      }
      acc[t] = wmma4(a, b, acc[t]);
    }
  }
}

// EPI: 0 = plain store, 1 = +bias[col], 2 = +I, 3 = 2I - acc   (Newton residual)
template <int AMODE, int BMODE, int EPI>
__global__ void k_gemm(const float* __restrict__ A, int lda,
                       const float* __restrict__ B, int ldb,
                       float* __restrict__ C, int ldc,
                       int Kd, int Mt, int cgN, const float* __restrict__ extra) {
  const int tile = blockIdx.x * (blockDim.x >> 5) + (threadIdx.x >> 5);
  if (tile >= Mt * cgN) return;
  const int m0 = (tile / cgN) * 16;
  const int n0 = (tile % cgN) * 64;
  v8f acc[4] = {};
  gemm_core<AMODE, BMODE>(A, lda, B, ldb, Kd, m0, n0, acc);
  const int lane = threadIdx.x & 31;
  const int r = lane & 15;
  const int half = lane >> 4;
#pragma unroll
  for (int t = 0; t < 4; ++t)
#pragma unroll
    for (int v = 0; v < 8; ++v) {
      const int row = m0 + v + 8 * half;
      const int col = n0 + 16 * t + r;
      float val = acc[t][v];
      if (EPI == 1) val += extra[col];
      if (EPI == 2) val += (row == col) ? 1.0f : 0.0f;
      if (EPI == 3) val = ((row == col) ? 2.0f : 0.0f) - val;
      C[(size_t)row * ldc + col] = val;
    }
}

// ---------------- small helpers ----------------
__device__ inline float blockReduceSum(float v, float* sbuf) {
  const int tid = threadIdx.x;
  sbuf[tid] = v;
  __syncthreads();
  for (int off = blockDim.x >> 1; off > 0; off >>= 1) {
    if (tid < off) sbuf[tid] += sbuf[tid + off];
    __syncthreads();
  }
  float r = sbuf[0];
  __syncthreads();
  return r;
}

__global__ void k_zero(float* p, int n) {
  int i = blockIdx.x * blockDim.x + threadIdx.x;
  if (i < n) p[i] = 0.0f;
}

// LayerNorm over D=256, in place. One block (256 thr) per row.
__global__ void k_layernorm(float* __restrict__ X, const float* __restrict__ gamma,
                            const float* __restrict__ beta) {
  __shared__ float sbuf[256];
  const int tid = threadIdx.x;
  const size_t row = blockIdx.x;
  float x = X[row * D_EMB + tid];
  float mean = blockReduceSum(x, sbuf) * (1.0f / D_EMB);
  float m2 = blockReduceSum(x * x, sbuf) * (1.0f / D_EMB);
  float var = m2 - mean * mean;
  X[row * D_EMB + tid] = gamma[tid] * ((x - mean) * rsqrtf(var + EPSV)) + beta[tid];
}

// cnorm[k] = sum_d C[d][k]^2 ; one block per k.
__global__ void k_colnorm(const float* __restrict__ Cw, float* __restrict__ cn) {
  __shared__ float sbuf[256];
  const int k = blockIdx.x;
  float v = Cw[(size_t)threadIdx.x * K_EMB + k];
  float s = blockReduceSum(v * v, sbuf);
  if (threadIdx.x == 0) cn[k] = s;
}

// argmin_k (cnorm[k] - 2 S[m][k]); then S[m][amin] += lambda  (turns S into b^T).
__global__ void k_argmin(float* __restrict__ S, const float* __restrict__ cn,
                         int* __restrict__ assign) {
  const int wid = blockIdx.x * (blockDim.x >> 5) + (threadIdx.x >> 5);
  if (wid >= M_TOK) return;
  const int lane = threadIdx.x & 31;
  float best = 3.4e38f;
  int bk = 0;
  for (int k = lane; k < K_EMB; k += 32) {
    float dv = cn[k] - 2.0f * S[(size_t)wid * K_EMB + k];
    if (dv < best) { best = dv; bk = k; }
  }
  for (int s = 16; s > 0; s >>= 1) {
    float ob = __shfl_xor(best, s, 32);
    int ok = __shfl_xor(bk, s, 32);
    if (ob < best || (ob == best && ok < bk)) { best = ob; bk = ok; }
  }
  if (lane == 0) {
    assign[wid] = bk;
    S[(size_t)wid * K_EMB + bk] += LAMBDA_REG;
  }
}

// rowsum[i] = sum_j |A[i][j]|  (for Newton-Schulz seed); one block per row.
__global__ void k_rowabs(const float* __restrict__ A, float* __restrict__ rs) {
  __shared__ float sbuf[256];
  const size_t row = blockIdx.x;
  float s = fabsf(A[row * K_EMB + threadIdx.x]) +
            fabsf(A[row * K_EMB + threadIdx.x + 256]);
  s = blockReduceSum(s, sbuf);
  if (threadIdx.x == 0) rs[row] = s;
}

__global__ void k_scalemax(const float* __restrict__ rs, float* __restrict__ scal) {
  __shared__ float sbuf[512];
  const int tid = threadIdx.x;
  sbuf[tid] = rs[tid];
  __syncthreads();
  for (int off = 256; off > 0; off >>= 1) {
    if (tid < off) sbuf[tid] = fmaxf(sbuf[tid], sbuf[tid + off]);
    __syncthreads();
  }
  if (tid == 0) scal[0] = 1.0f / (sbuf[0] * sbuf[0]);  // 1/(||A||1 * ||A||inf)
}

__global__ void k_scaleA(const float* __restrict__ A, const float* __restrict__ scal,
                         float* __restrict__ X0) {
  int i = blockIdx.x * blockDim.x + threadIdx.x;
  X0[i] = scal[0] * A[i];
}

// Simplex projection per token: bitonic sort (desc) + inclusive scan in LDS.
__global__ void k_simplex(const float* __restrict__ Psol, float* __restrict__ Pout) {
  __shared__ float sv[512];
  __shared__ float sc[512];
  __shared__ int si[512];
  const int tid = threadIdx.x;
  const size_t row = blockIdx.x;
  const float orig = Psol[row * K_EMB + tid];
  sv[tid] = orig;
  __syncthreads();
  for (int size = 2; size <= 512; size <<= 1) {
    for (int stride = size >> 1; stride > 0; stride >>= 1) {
      const int pos = tid ^ stride;
      const float a = sv[tid];
      const float b = sv[pos];
      const bool desc = ((tid & size) == 0);
      if (pos > tid) {
        const bool sw = desc ? (a < b) : (a > b);
        if (sw) { sv[tid] = b; sv[pos] = a; }
      }
      __syncthreads();
    }
  }
  sc[tid] = sv[tid];
  __syncthreads();
  for (int off = 1; off < 512; off <<= 1) {
    float x = sc[tid];
    float add = (tid >= off) ? sc[tid - off] : 0.0f;
    __syncthreads();
    sc[tid] = x + add;
    __syncthreads();
  }
  const float thr = (sc[tid] - 1.0f) / (float)(tid + 1);
  si[tid] = (sv[tid] > thr) ? tid : 0;
  __syncthreads();
  for (int off = 256; off > 0; off >>= 1) {
    if (tid < off) si[tid] = max(si[tid], si[tid + off]);
    __syncthreads();
  }
  const int rho = si[0];
  const float theta = (sc[rho] - 1.0f) / (float)(rho + 1);
  Pout[row * K_EMB + tid] = fmaxf(orig - theta, 0.0f);
}

// psum[k] += sum over a 128-row chunk of P[m][k]; 512 thr = one k each, coalesced.
__global__ void k_psum(const float* __restrict__ P, float* __restrict__ psum) {
  const int k = threadIdx.x;
  const size_t r0 = (size_t)blockIdx.x * 128;
  float s = 0.0f;
  for (int i = 0; i < 128; ++i) s += P[(r0 + i) * K_EMB + k];
  atomicAdd(&psum[k], s);
}

__global__ void k_sse(const float* __restrict__ zq, const float* __restrict__ flat,
                      float* __restrict__ acc) {
  __shared__ float sbuf[256];
  const size_t n = (size_t)M_TOK * D_EMB;
  const size_t stride = (size_t)gridDim.x * blockDim.x;
  float s = 0.0f;
  for (size_t i = (size_t)blockIdx.x * blockDim.x + threadIdx.x; i < n; i += stride) {
    float d = zq[i] - flat[i];
    s += d * d;
  }
  s = blockReduceSum(s, sbuf);
  if (threadIdx.x == 0) atomicAdd(acc, s);
}

// diversity: sum_{i!=j} relu(G_ij / (||ci|| ||cj||)), G = A - I.
__global__ void k_div(const float* __restrict__ A, const float* __restrict__ cn,
                      float* __restrict__ acc) {
  __shared__ float sbuf[256];
  const int n = K_EMB * K_EMB;
  const int stride = gridDim.x * blockDim.x;
  float s = 0.0f;
  for (int idx = blockIdx.x * blockDim.x + threadIdx.x; idx < n; idx += stride) {
    const int i = idx / K_EMB, j = idx % K_EMB;
    if (i != j) {
      float v = A[idx] * rsqrtf(fmaxf(cn[i], 1e-12f)) * rsqrtf(fmaxf(cn[j], 1e-12f));
      s += fmaxf(v, 0.0f);
    }
  }
  s = blockReduceSum(s, sbuf);
  if (threadIdx.x == 0) atomicAdd(acc, s);
}

__global__ void k_finalize(const float* __restrict__ psum, const float* __restrict__ scal,
                           float* __restrict__ out2) {
  __shared__ float sbuf[512];
  const int tid = threadIdx.x;
  const float ps = psum[tid];
  const float Minv = 1.0f / (float)M_TOK;
  // perplexity
  float pj = fminf(fmaxf(ps * Minv, 1e-10f), 1.0f - 1e-9f);
  float Spj = blockReduceSum(pj, sbuf);
  float pjn = pj / Spj;
  float ent = blockReduceSum(-pjn * log2f(pjn), sbuf);
  // entropy regularizer
  float pm = ps * Minv;
  float Spm = blockReduceSum(pm, sbuf);
  float pmn = pm / (Spm + EPSV);
  float ereg = blockReduceSum(-pmn * logf(pmn + EPSV), sbuf);
  if (tid == 0) {
    float primary = scal[1] / (float)((size_t)M_TOK * D_EMB);
    float diversity = scal[2] / (float)((size_t)K_EMB * K_EMB);
    out2[0] = primary + A_ENT * ereg + A_DIV * diversity;  // loss
    out2[1] = exp2f(ent);                                   // perplexity
  }
}

// ---------------- driver ----------------
extern "C" void kernel_launch(void* const* d_in, const int* in_sizes, int n_in,
                              void* d_out, int out_size, void* d_ws, size_t ws_size,
                              hipStream_t stream) {
  const float* inputs = (const float*)d_in[0];
  const float* gamma  = (const float*)d_in[1];
  const float* beta   = (const float*)d_in[2];
  const float* embed  = (const float*)d_in[3];  // (D,K) row-major
  const float* dw     = (const float*)d_in[4];  // (D,D)
  const float* db     = (const float*)d_in[5];  // (D,)

  float* out  = (float*)d_out;
  float* zq   = out;                                   // M*D
  float* pout = out + (size_t)M_TOK * D_EMB;           // M*K
  float* out2 = pout + (size_t)M_TOK * K_EMB;          // loss, perplexity

  float* ws = (float*)d_ws;
  size_t off = 0;
  float* flat = ws + off; off += (size_t)M_TOK * D_EMB;
  float* S    = ws + off; off += (size_t)M_TOK * K_EMB;   // dists GEMM -> b^T
  float* Psol = ws + off; off += (size_t)M_TOK * K_EMB;
  float* Amat = ws + off; off += (size_t)K_EMB * K_EMB;
  float* X0   = ws + off; off += (size_t)K_EMB * K_EMB;
  float* X1   = ws + off; off += (size_t)K_EMB * K_EMB;
  float* Rm   = ws + off; off += (size_t)K_EMB * K_EMB;
  float* cn   = ws + off; off += K_EMB;
  float* rs   = ws + off; off += K_EMB;
  float* psum = ws + off; off += K_EMB;
  float* scal = ws + off; off += 8;   // [0]=NS seed scale, [1]=sse, [2]=div
  int* assign = (int*)(ws + off);

  // zero accumulators (psum + scal are contiguous)
  k_zero<<<2, 512, 0, stream>>>(psum, K_EMB + 8);

  // dense + bias -> layernorm (in place) -> flat
  k_gemm<0, 0, 1><<<(M_TOK / 16) * (D_EMB / 64) / 8, 256, 0, stream>>>(
      inputs, D_EMB, dw, D_EMB, flat, D_EMB, D_EMB, M_TOK / 16, D_EMB / 64, db);
  k_layernorm<<<M_TOK, 256, 0, stream>>>(flat, gamma, beta);

  // codebook column norms; S = flat @ C  (serves dists and b^T)
  k_colnorm<<<K_EMB, 256, 0, stream>>>(embed, cn);
  k_gemm<0, 0, 0><<<(M_TOK / 16) * (K_EMB / 64) / 8, 256, 0, stream>>>(
      flat, D_EMB, embed, K_EMB, S, K_EMB, D_EMB, M_TOK / 16, K_EMB / 64, nullptr);
  k_argmin<<<M_TOK / 8, 256, 0, stream>>>(S, cn, assign);

  // A = C^T C + I
  k_gemm<1, 0, 2><<<(K_EMB / 16) * (K_EMB / 64) / 8, 256, 0, stream>>>(
      embed, K_EMB, embed, K_EMB, Amat, K_EMB, D_EMB, K_EMB / 16, K_EMB / 64, nullptr);

  // Newton-Schulz inverse: X0 = A / (||A||1 ||A||inf); X <- X(2I - AX)
  k_rowabs<<<K_EMB, 256, 0, stream>>>(Amat, rs);
  k_scalemax<<<1, 512, 0, stream>>>(rs, scal);
  k_scaleA<<<(K_EMB * K_EMB) / 256, 256, 0, stream>>>(Amat, scal, X0);
  float* Xc = X0;
  float* Xn = X1;
  for (int it = 0; it < NS_ITERS; ++it) {
    k_gemm<0, 0, 3><<<(K_EMB / 16) * (K_EMB / 64) / 8, 256, 0, stream>>>(
        Xc, K_EMB, Amat, K_EMB, Rm, K_EMB, K_EMB, K_EMB / 16, K_EMB / 64, nullptr);
    k_gemm<0, 0, 0><<<(K_EMB / 16) * (K_EMB / 64) / 8, 256, 0, stream>>>(
        Xc, K_EMB, Rm, K_EMB, Xn, K_EMB, K_EMB, K_EMB / 16, K_EMB / 64, nullptr);
    float* t = Xc; Xc = Xn; Xn = t;
  }

  // P_sol^T = b^T @ Ainv (Ainv symmetric)
  k_gemm<0, 0, 0><<<(M_TOK / 16) * (K_EMB / 64) / 8, 256, 0, stream>>>(
      S, K_EMB, Xc, K_EMB, Psol, K_EMB, K_EMB, M_TOK / 16, K_EMB / 64, nullptr);

  // simplex projection -> out_P_proj region of d_out; per-code sums
  k_simplex<<<M_TOK, 512, 0, stream>>>(Psol, pout);
  k_psum<<<M_TOK / 128, 512, 0, stream>>>(pout, psum);

  // Zq = P @ C^T  -> d_out
  k_gemm<0, 1, 0><<<(M_TOK / 16) * (D_EMB / 64) / 8, 256, 0, stream>>>(
      pout, K_EMB, embed, K_EMB, zq, D_EMB, K_EMB, M_TOK / 16, D_EMB / 64, nullptr);

  // scalar losses
  k_sse<<<4096, 256, 0, stream>>>(zq, flat, scal + 1);
  k_div<<<256, 256, 0, stream>>>(Amat, cn, scal + 2);
  k_finalize<<<1, 512, 0, stream>>>(psum, scal, out2);
}